// MultiGrid_60816736911552
// MI455X (gfx1250) — compile-verified
//
#include <hip/hip_runtime.h>

// ---------------------------------------------------------------------------
// Batched multigrid V-cycle (B=8, 1025^2, 10 levels, 6 cycles), f32.
// Memory-bound: fused kernels, working set (~130MB) is L2-resident (192MB).
// Smoother: conv3(v,K) = 3v - box9(v)/3; box9 row-sum over a 16x16 tile is a
// 16x16x16 f32 matmul with a constant 0/1 tridiagonal B -> 4 chained
// v_wmma_f32_16x16x4_f32 (exact in f32). Halo tiles are DMA'd into LDS with
// the Tensor Data Mover (tensor_load_to_lds, TENSORcnt) on interior tiles.
// ---------------------------------------------------------------------------

typedef float v2f __attribute__((ext_vector_type(2)));
typedef float v8f __attribute__((ext_vector_type(8)));
typedef unsigned int u32x4 __attribute__((ext_vector_type(4)));
typedef int i32x4 __attribute__((ext_vector_type(4)));
typedef int i32x8 __attribute__((ext_vector_type(8)));

#define NB 8
#define NLEV 10
#define VH_W 18                 // contiguous 18-float rows (TDM 2D tile dest)
#define VH_ELEMS (18 * VH_W)
#define CS_W 20                 // 16x18 column-sum tile, padded stride
#define CS_ELEMS (16 * CS_W)

#if defined(__gfx1250__) && __has_builtin(__builtin_amdgcn_tensor_load_to_lds)
#define USE_TDM 1
#endif

#ifdef USE_TDM
// One-shot 2D tile DMA: 18x18 f32 tile from global (row stride = stride_elems)
// into contiguous LDS at lds_dst. D# per CDNA5 ISA 8.3/8.4; groups 2/3 unused.
__device__ __forceinline__ void tdm_load_tile_18x18(const float* gsrc,
                                                    float* lds_dst,
                                                    int stride_elems) {
  unsigned long long ga = (unsigned long long)(size_t)gsrc;
  unsigned int laddr = (unsigned int)(size_t)lds_dst;  // LDS offset = addr[31:0]
  u32x4 g0;
  g0.x = 1u;                                            // count=1 (valid D#)
  g0.y = laddr;                                         // lds_addr
  g0.z = (unsigned int)ga;                              // global_addr[31:0]
  g0.w = (unsigned int)((ga >> 32) & 0x1FFFFFFull) | 0x80000000u;  // [56:32]|type=2
  i32x8 g1;
  g1[0] = 0x20000;            // workgroup_mask=0, data_size=2 (4 bytes)
  g1[1] = 18 << 16;           // tensor_dim0[15:0]=18 (tile fully in-bounds)
  g1[2] = 18 << 16;           // tensor_dim0 hi=0 | tensor_dim1[15:0]=18
  g1[3] = 18 << 16;           // tensor_dim1 hi=0 | tile_dim0=18
  g1[4] = 18;                 // tile_dim1=18, tile_dim2=0
  g1[5] = stride_elems;       // tensor_dim0_stride[31:0]
  g1[6] = 0;
  g1[7] = 0;
  i32x4 gz = {0, 0, 0, 0};
#if __clang_major__ >= 23
  i32x8 gz8 = {0, 0, 0, 0, 0, 0, 0, 0};
  __builtin_amdgcn_tensor_load_to_lds(g0, g1, gz, gz, gz8, 0);
#else
  __builtin_amdgcn_tensor_load_to_lds(g0, g1, gz, gz, 0);
#endif
}
#endif

// ---------------- FNet: f = conv3(F, MASS), MASS = (h^2/36)[1,4,1]^T[1,4,1]
__global__ void k_mass(const float* __restrict__ F, float* __restrict__ out) {
  const int dim = 1025;
  const int plane = dim * dim;
  int idx = blockIdx.x * blockDim.x + threadIdx.x;
  if (idx >= NB * plane) return;
  int b = idx / plane, r = idx % plane;
  int y = r / dim, x = r % dim;
  const float* Fb = F + (long)b * plane;
  const float wr[3] = {1.f, 4.f, 1.f};
  float s = 0.f;
#pragma unroll
  for (int a = 0; a < 3; a++) {
    int yy = y + a - 1;
    if (yy < 0 || yy >= dim) continue;
#pragma unroll
    for (int c = 0; c < 3; c++) {
      int xx = x + c - 1;
      if (xx < 0 || xx >= dim) continue;
      s += wr[a] * wr[c] * Fb[yy * dim + xx];
    }
  }
  out[idx] = s * 1.0596381293402778e-07f;  // (2/1024)^2 / 36
}

// ---------------- shared WMMA jacobi tile epilogue ----------------
// VH: 18x18 halo tile of the smoother input u (VH[r][c] = u(tile_r0-1+r, ...)).
// Computes v' = (u + 0.25*(f - (3u - box9(u)/3))) * interior_mask and stores.
__device__ __forceinline__ void jacobi_tile_wmma(
    float* VH, float* CS, int lane, int dim, int tr0, int tc0, long bplane,
    const float* __restrict__ f, float* __restrict__ vout, bool active) {
  __syncthreads();  // VH visible
  // cooperative column sums: CS[m][c] = sum of VH rows m..m+2 at col c (c=0..17)
  for (int i = lane; i < 16 * 18; i += 32) {
    int m = i / 18, c = i % 18;
    CS[m * CS_W + c] =
        VH[m * VH_W + c] + VH[(m + 1) * VH_W + c] + VH[(m + 2) * VH_W + c];
  }
  __syncthreads();
  const int half = lane >> 4;
  const int l15  = lane & 15;
  // D/C layout: VGPR r <-> (M = r + 8*half, N = l15).
  // A layout:   lane -> row M=l15; VGPR t, half h -> K = 4k + 2h + t.
  // B layout:   lane -> col N=l15; VGPR t, half h -> K = 4k + 2h + t.
  v8f acc = {0.f, 0.f, 0.f, 0.f, 0.f, 0.f, 0.f, 0.f};
#pragma unroll
  for (int k = 0; k < 4; ++k) {
    const int K0 = 4 * k + 2 * half;
    v2f a, bm;
    a.x = CS[l15 * CS_W + K0 + 1];  // CS col index = K+1 (col 0 is halo -1)
    a.y = CS[l15 * CS_W + K0 + 2];
    int d0 = K0 - l15, d1 = K0 + 1 - l15;
    bm.x = (d0 >= -1 && d0 <= 1) ? 1.f : 0.f;  // tridiagonal row-sum matrix
    bm.y = (d1 >= -1 && d1 <= 1) ? 1.f : 0.f;
    acc = __builtin_amdgcn_wmma_f32_16x16x4_f32(false, a, false, bm, (short)0,
                                                acc, false, false);
  }
#pragma unroll
  for (int r = 0; r < 8; ++r) {
    int row = r + 8 * half;
    float box = acc[r];
    if (l15 == 0)  box += CS[row * CS_W + 0];   // halo column -1
    if (l15 == 15) box += CS[row * CS_W + 17];  // halo column 16
    float vc = VH[(row + 1) * VH_W + (l15 + 1)];
    float kv = 3.f * vc - box * (1.f / 3.f);    // == conv3(u, K_STENCIL)
    int gr = tr0 + row, gc = tc0 + l15;
    if (active && gr < dim && gc < dim) {
      float fv = f[bplane + (long)gr * dim + gc];
      float o = vc + 0.25f * (fv - kv);         // omega/D_center = 0.25 exactly
      if (gr == 0 || gc == 0 || gr == dim - 1 || gc == dim - 1) o = 0.f;
      vout[bplane + (long)gr * dim + gc] = o;
    }
  }
}

// ---------------- weighted-Jacobi sweep: vout = jacobi(vin, f) ----------------
__global__ void k_jacobi(int dim, const float* __restrict__ vin,
                         const float* __restrict__ f, float* __restrict__ vout) {
  __shared__ float sVH[8 * VH_ELEMS];
  __shared__ float sCS[8 * CS_ELEMS];
  int lane = threadIdx.x & 31;
  int wave = threadIdx.x >> 5;
  float* VH = sVH + wave * VH_ELEMS;
  float* CS = sCS + wave * CS_ELEMS;
  int tpd = (dim + 15) >> 4;
  int tilesTotal = NB * tpd * tpd;
  int wid = blockIdx.x * 8 + wave;
  bool active = wid < tilesTotal;
  int tile = active ? wid : 0;
  int b  = tile / (tpd * tpd);
  int tr = (tile % (tpd * tpd)) / tpd;
  int tc = tile % tpd;
  int tr0 = tr * 16, tc0 = tc * 16;
  long bplane = (long)b * dim * dim;
  const float* vb = vin + bplane;
  __builtin_prefetch(&f[bplane + (long)tr0 * dim + tc0], 0, 3);
#ifdef USE_TDM
  // interior tile: whole 18x18 halo window inside the grid -> one TDM DMA
  bool interior = (tr0 >= 1) && (tc0 >= 1) &&
                  (tr0 + 16 <= dim - 1) && (tc0 + 16 <= dim - 1);
  if (interior) {
    tdm_load_tile_18x18(vb + (long)(tr0 - 1) * dim + (tc0 - 1), VH, dim);
  } else {
    for (int i = lane; i < 18 * 18; i += 32) {
      int r = i / 18, c = i % 18;
      int gr = tr0 + r - 1, gc = tc0 + c - 1;
      float val = 0.f;
      if (gr >= 0 && gr < dim && gc >= 0 && gc < dim)
        val = vb[(long)gr * dim + gc];
      VH[r * VH_W + c] = val;
    }
  }
  __builtin_amdgcn_s_wait_tensorcnt(0);
#else
  for (int i = lane; i < 18 * 18; i += 32) {
    int r = i / 18, c = i % 18;
    int gr = tr0 + r - 1, gc = tc0 + c - 1;
    float val = 0.f;
    if (gr >= 0 && gr < dim && gc >= 0 && gc < dim)
      val = vb[(long)gr * dim + gc];
    VH[r * VH_W + c] = val;
  }
#endif
  jacobi_tile_wmma(VH, CS, lane, dim, tr0, tc0, bplane, f, vout, active);
}

// ---------------- fused residual + full-weighting restrict + jacobi(0,fc) ----
// fc(ic,jc) = w0 * sum_{a,b} R[a,b] * (ff - K*vf)(2ic-1+a, 2jc-1+b), borders 0.
// vc = jacobi(0, fc) = 0.25 * fc.
__global__ void k_restrict(int dimf, int dimc, const float* __restrict__ vf,
                           const float* __restrict__ ff, float* __restrict__ fc,
                           float* __restrict__ vc, const float* __restrict__ R,
                           const float* __restrict__ w) {
  int planec = dimc * dimc;
  int idx = blockIdx.x * blockDim.x + threadIdx.x;
  if (idx >= NB * planec) return;
  int b = idx / planec, rr = idx % planec;
  int ic = rr / dimc, jc = rr % dimc;
  float acc = 0.f;
  if (ic > 0 && ic < dimc - 1 && jc > 0 && jc < dimc - 1) {
    const float w0 = w[0];
    const float* vb = vf + (long)b * dimf * dimf;
    const float* fb = ff + (long)b * dimf * dimf;
#pragma unroll
    for (int a = 0; a < 3; a++) {
      int x = 2 * ic - 1 + a;  // fine row, always interior
#pragma unroll
      for (int c = 0; c < 3; c++) {
        int y = 2 * jc - 1 + c;  // fine col, always interior
        float ctr = vb[(long)x * dimf + y];
        float s8 = vb[(long)(x - 1) * dimf + y - 1] + vb[(long)(x - 1) * dimf + y] +
                   vb[(long)(x - 1) * dimf + y + 1] + vb[(long)x * dimf + y - 1] +
                   vb[(long)x * dimf + y + 1] + vb[(long)(x + 1) * dimf + y - 1] +
                   vb[(long)(x + 1) * dimf + y] + vb[(long)(x + 1) * dimf + y + 1];
        float kv = (8.f * ctr - s8) * (1.f / 3.f);
        acc += R[a * 3 + c] * (fb[(long)x * dimf + y] - kv);
      }
    }
    acc *= w0;
  }
  fc[idx] = acc;
  vc[idx] = 0.25f * acc;  // jacobi on zero initial guess
}

// ---------------- fused prolong (convT k3 s2 p1 with flip(P)) + correct + jacobi
__global__ void k_prolong(int dim, int dimc, const float* __restrict__ vcrs,
                          const float* __restrict__ vin,
                          const float* __restrict__ f,
                          const float* __restrict__ P,
                          const float* __restrict__ w, float* __restrict__ vout) {
  __shared__ float sVH[8 * VH_ELEMS];
  __shared__ float sCS[8 * CS_ELEMS];
  int lane = threadIdx.x & 31;
  int wave = threadIdx.x >> 5;
  float* VH = sVH + wave * VH_ELEMS;
  float* CS = sCS + wave * CS_ELEMS;
  int tpd = (dim + 15) >> 4;
  int tilesTotal = NB * tpd * tpd;
  int wid = blockIdx.x * 8 + wave;
  bool active = wid < tilesTotal;
  int tile = active ? wid : 0;
  int b  = tile / (tpd * tpd);
  int tr = (tile % (tpd * tpd)) / tpd;
  int tc = tile % tpd;
  int tr0 = tr * 16, tc0 = tc * 16;
  long bplane = (long)b * dim * dim;
  const float* vb = vin + bplane;
  const float* cb = vcrs + (long)b * dimc * dimc;
  const float w1 = w[1];
  float Pl[9];
#pragma unroll
  for (int i = 0; i < 9; i++) Pl[i] = P[i];
  // fill halo tile with u = vin + w1 * e, e = convT(vcrs, flip(P), s2, p1)
  for (int i = lane; i < 18 * 18; i += 32) {
    int r = i / 18, c = i % 18;
    int gx = tr0 + r - 1, gy = tc0 + c - 1;
    float val = 0.f;
    if (gx >= 0 && gx < dim && gy >= 0 && gy < dim) {
      float e;
      int i0 = gx >> 1, j0 = gy >> 1;
      if ((gx & 1) == 0) {
        if ((gy & 1) == 0) {
          e = Pl[4] * cb[(long)i0 * dimc + j0];
        } else {
          e = Pl[5] * cb[(long)i0 * dimc + j0] +
              Pl[3] * cb[(long)i0 * dimc + j0 + 1];
        }
      } else {
        if ((gy & 1) == 0) {
          e = Pl[7] * cb[(long)i0 * dimc + j0] +
              Pl[1] * cb[(long)(i0 + 1) * dimc + j0];
        } else {
          e = Pl[8] * cb[(long)i0 * dimc + j0] +
              Pl[6] * cb[(long)i0 * dimc + j0 + 1] +
              Pl[2] * cb[(long)(i0 + 1) * dimc + j0] +
              Pl[0] * cb[(long)(i0 + 1) * dimc + j0 + 1];
        }
      }
      val = vb[(long)gx * dim + gy] + w1 * e;
    }
    VH[r * VH_W + c] = val;
  }
  jacobi_tile_wmma(VH, CS, lane, dim, tr0, tc0, bplane, f, vout, active);
}

// ---------------------------------------------------------------------------
extern "C" void kernel_launch(void* const* d_in, const int* in_sizes, int n_in,
                              void* d_out, int out_size, void* d_ws,
                              size_t ws_size, hipStream_t stream) {
  (void)in_sizes; (void)n_in; (void)out_size; (void)ws_size;
  const float* F  = (const float*)d_in[0];
  const float* v0 = (const float*)d_in[1];
  const float* R  = (const float*)d_in[2];
  const float* P  = (const float*)d_in[3];
  const float* w  = (const float*)d_in[4];
  float* out = (float*)d_out;

  int dims[NLEV];
  for (int j = 0; j < NLEV; j++) dims[j] = (1024 >> j) + 1;

  // workspace layout (~96 MB of f32): fs[0..9], one level-0 v buffer
  // (d_out is the other), ping-pong pair per coarse level.
  float* wsf = (float*)d_ws;
  size_t off = 0;
  float* fs[NLEV];
  for (int j = 0; j < NLEV; j++) {
    fs[j] = wsf + off;
    off += (size_t)NB * dims[j] * dims[j];
  }
  float* v0a = wsf + off;
  off += (size_t)NB * dims[0] * dims[0];
  float* pa[NLEV];
  float* pb[NLEV];
  for (int j = 1; j < NLEV; j++) {
    pa[j] = wsf + off; off += (size_t)NB * dims[j] * dims[j];
    pb[j] = wsf + off; off += (size_t)NB * dims[j] * dims[j];
  }

  // f = conv3(F, MASS), once
  {
    int n = NB * dims[0] * dims[0];
    k_mass<<<(n + 255) / 256, 256, 0, stream>>>(F, fs[0]);
  }

  float* lvl0buf[2] = {v0a, out};  // 12 alternating writes -> last lands in out
  int t0 = 0;
  const float* cur0 = v0;
  const float* cur[NLEV];

  for (int cyc = 0; cyc < 6; ++cyc) {
    {  // pre-smooth level 0
      float* dst = lvl0buf[t0]; t0 ^= 1;
      int tpd = (dims[0] + 15) / 16;
      int waves = NB * tpd * tpd;
      k_jacobi<<<(waves + 7) / 8, 256, 0, stream>>>(dims[0], cur0, fs[0], dst);
      cur0 = dst;
    }
    // downward: residual -> restrict -> relax(0)
    for (int j = 0; j < NLEV - 1; j++) {
      const float* vf = (j == 0) ? cur0 : cur[j];
      int dimc = dims[j + 1];
      int n = NB * dimc * dimc;
      k_restrict<<<(n + 255) / 256, 256, 0, stream>>>(dims[j], dimc, vf, fs[j],
                                                      fs[j + 1], pa[j + 1], R, w);
      cur[j + 1] = pa[j + 1];
    }
    {  // extra smooth at coarsest level
      int dim = dims[NLEV - 1];
      int tpd = (dim + 15) / 16;
      int waves = NB * tpd * tpd;
      k_jacobi<<<(waves + 7) / 8, 256, 0, stream>>>(dim, cur[NLEV - 1],
                                                    fs[NLEV - 1], pb[NLEV - 1]);
      cur[NLEV - 1] = pb[NLEV - 1];
    }
    // upward: prolong -> correct -> smooth
    for (int j = NLEV - 2; j >= 0; j--) {
      int dim = dims[j], dimc = dims[j + 1];
      int tpd = (dim + 15) / 16;
      int waves = NB * tpd * tpd;
      const float* vj = (j == 0) ? cur0 : cur[j];
      float* dst;
      if (j == 0) { dst = lvl0buf[t0]; t0 ^= 1; } else { dst = pb[j]; }
      k_prolong<<<(waves + 7) / 8, 256, 0, stream>>>(dim, dimc, cur[j + 1], vj,
                                                     fs[j], P, w, dst);
      if (j == 0) cur0 = dst; else cur[j] = dst;
    }
  }
  // by construction the 12th level-0 write (final smooth) landed in d_out
}